// APPM_8031588843744
// MI455X (gfx1250) — compile-verified
//
#include <hip/hip_runtime.h>

typedef float v2f __attribute__((ext_vector_type(2)));
typedef float v8f __attribute__((ext_vector_type(8)));

#define FM        14
#define NPOS      196      // 14*14
#define NCH       2048
#define NBATCH    64
#define NWINS     837
#define NRATIO    11
#define NPTILE    13       // ceil(196/16)

__constant__ int c_rh[NRATIO]     = {4,3,5,6,5,7,8,6,10,7,9};
__constant__ int c_rw[NRATIO]     = {4,5,3,6,7,5,8,10,6,9,7};
__constant__ int c_wbase[NRATIO+1]= {0,121,241,361,442,522,602,651,696,741,789,837};

// ---------------------------------------------------------------------------
// Kernel 1: channel reduction y[b,p] = sum_c x[b,c,p] via V_WMMA_F32_16X16X4_F32
// A = all-ones 16x4  =>  D[0,n] = sum_k B[k,n]  (column sums of B)
// Each wave: 16 positions x 256 channels; 8 waves/block cover all 2048 channels.
// Unroll-8 keeps a 16-load s_clause in flight per wave for HBM latency hiding.
// ---------------------------------------------------------------------------
__global__ __launch_bounds__(256) void k_chansum_wmma(const float* __restrict__ x,
                                                      float* __restrict__ y) {
    const int bx   = blockIdx.x;
    const int b    = bx / NPTILE;
    const int pt   = bx % NPTILE;
    const int p0   = pt * 16;

    const int tid  = threadIdx.x;
    const int wave = tid >> 5;       // 0..7 -> channel chunk of 256
    const int lane = tid & 31;
    const int half = lane >> 4;      // B-matrix K rows {0,1} vs {2,3}
    const int l    = lane & 15;      // column n (position)

    const int p  = p0 + l;
    const int pc = p < NPOS ? p : NPOS - 1;   // clamped tail (dupes never stored)

    const int cstart = wave * 256;
    const float* ptr = x + ((size_t)b * NCH + (size_t)(cstart + 2 * half)) * NPOS + pc;

    v2f a; a.x = 1.0f; a.y = 1.0f;            // all-ones A matrix
    v8f acc = {};

    #pragma unroll 8
    for (int k = 0; k < 64; ++k) {            // 64 * 4 = 256 channels
        v2f bm;
        bm.x = ptr[0];                        // channel c + 2*half
        bm.y = ptr[NPOS];                     // channel c + 2*half + 1
        acc = __builtin_amdgcn_wmma_f32_16x16x4_f32(false, a, false, bm,
                                                    (short)0, acc, false, false);
        ptr += 4 * NPOS;
    }

    // D row M=0 lives in acc[0] of lanes 0..15 (N = lane)
    __shared__ float lds[8 * 16];
    if (half == 0) lds[wave * 16 + l] = acc[0];
    __syncthreads();
    if (tid < 16) {
        float s = 0.f;
        #pragma unroll
        for (int w = 0; w < 8; ++w) s += lds[w * 16 + tid];
        const int pp = p0 + tid;
        if (pp < NPOS) y[b * NPOS + pp] = s;
    }
}

// ---------------------------------------------------------------------------
// Kernel 2: 837 window mean-scores per batch via 15x15 integral image in LDS.
// ---------------------------------------------------------------------------
__global__ __launch_bounds__(256) void k_window_scores(const float* __restrict__ y,
                                                       float* __restrict__ all_sc) {
    const int b   = blockIdx.x;
    const int tid = threadIdx.x;

    __shared__ float yt[FM][FM];
    __shared__ float S[FM + 1][FM + 1];

    if (tid < NPOS) yt[tid / FM][tid % FM] = y[b * NPOS + tid];
    if (tid < FM + 1) { S[0][tid] = 0.f; S[tid][0] = 0.f; }
    __syncthreads();

    if (tid < FM) {                      // row prefix sums
        float s = 0.f;
        for (int j = 0; j < FM; ++j) { s += yt[tid][j]; S[tid + 1][j + 1] = s; }
    }
    __syncthreads();
    if (tid < FM) {                      // column prefix sums (in place)
        float s = 0.f;
        for (int i = 1; i <= FM; ++i) { s += S[i][tid + 1]; S[i][tid + 1] = s; }
    }
    __syncthreads();

    for (int w = tid; w < NWINS; w += blockDim.x) {
        int r = 0;
        while (w >= c_wbase[r + 1]) ++r;
        const int local = w - c_wbase[r];
        const int rh = c_rh[r], rw = c_rw[r];
        const int cols = FM - rw + 1;
        const int i = local / cols, j = local % cols;
        const float s = S[i + rh][j + rw] - S[i][j + rw] - S[i + rh][j] + S[i][j];
        all_sc[b * NWINS + w] = s / (float)(rh * rw);
    }
}

// ---------------------------------------------------------------------------
// Kernel 3: greedy NMS, one wave32 per (batch, group).
// ---------------------------------------------------------------------------
__device__ __forceinline__ void win_coords(int gw, float& x1, float& y1,
                                           float& x2, float& y2) {
    int r = 0;
    while (gw >= c_wbase[r + 1]) ++r;
    const int local = gw - c_wbase[r];
    const int rh = c_rh[r], rw = c_rw[r];
    const int cols = FM - rw + 1;
    const int i = local / cols, j = local % cols;
    x1 = (float)(j * 32);
    y1 = (float)(i * 32);
    x2 = (float)((j + rw) * 32 - 1);
    y2 = (float)((i + rh) * 32 - 1);
}

__global__ __launch_bounds__(32) void k_nms(const float* __restrict__ all_sc,
                                            float* __restrict__ out_idx,
                                            float* __restrict__ out_score) {
    const int task = blockIdx.x;
    const int b = task / 3;
    const int g = task % 3;

    int off, W, nsel, slot;
    if (g == 0)      { off = 0;   W = 361; nsel = 3; slot = 0; }
    else if (g == 1) { off = 361; W = 241; nsel = 2; slot = 3; }
    else             { off = 602; W = 235; nsel = 1; slot = 5; }
    const float thresh = 0.25f;
    const float NEGINF = -__builtin_inff();

    const int lane = threadIdx.x;

    float sc[12];
    unsigned valid = 0u;
    #pragma unroll
    for (int t = 0; t < 12; ++t) {
        const int w = lane + 32 * t;
        if (w < W) { sc[t] = all_sc[b * NWINS + off + w]; valid |= (1u << t); }
        else sc[t] = NEGINF;
    }

    int last = 0;
    for (int k = 0; k < nsel; ++k) {
        // lane-local first-max among valid windows (t ascending => lowest index wins ties)
        float bv = NEGINF; int bi = 0x7fffffff;
        #pragma unroll
        for (int t = 0; t < 12; ++t) {
            const int w = lane + 32 * t;
            if (w < W && ((valid >> t) & 1u) && sc[t] > bv) { bv = sc[t]; bi = w; }
        }
        const bool anyv = (__any((int)(valid != 0u)) != 0);

        // wave argmax butterfly: higher score wins, tie -> smaller index
        for (int d = 16; d >= 1; d >>= 1) {
            const float ov = __shfl_xor(bv, d, 32);
            const int   oi = __shfl_xor(bi, d, 32);
            if (ov > bv || (ov == bv && oi < bi)) { bv = ov; bi = oi; }
        }
        const int idx = anyv ? bi : last;
        last = idx;

        float sx1, sy1, sx2, sy2;
        win_coords(off + idx, sx1, sy1, sx2, sy2);
        const float sarea = (sx2 - sx1 + 1.f) * (sy2 - sy1 + 1.f);

        if (anyv) {
            #pragma unroll
            for (int t = 0; t < 12; ++t) {
                const int w = lane + 32 * t;
                if (w >= W) continue;
                float x1, y1, x2, y2;
                win_coords(off + w, x1, y1, x2, y2);
                const float area = (x2 - x1 + 1.f) * (y2 - y1 + 1.f);
                const float ltx = fmaxf(x1, sx1), lty = fmaxf(y1, sy1);
                const float rbx = fminf(x2, sx2), rby = fminf(y2, sy2);
                const float wx = rbx - ltx + 1.f, wy = rby - lty + 1.f;
                const float inter = (wx < 0.f || wy < 0.f) ? 0.f : wx * wy;
                const float iou = inter / (area + sarea - inter);
                if (iou > thresh || w == idx) valid &= ~(1u << t);
            }
        }

        if (lane == 0) {
            out_idx  [b * 6 + slot + k] = (float)(idx + off);
            out_score[b * 6 + slot + k] = all_sc[b * NWINS + off + idx];
        }
    }
}

// ---------------------------------------------------------------------------
extern "C" void kernel_launch(void* const* d_in, const int* in_sizes, int n_in,
                              void* d_out, int out_size, void* d_ws, size_t ws_size,
                              hipStream_t stream) {
    const float* x = (const float*)d_in[0];
    // d_in[1] = proposalN (6), fixed by N_LIST -> unused
    float* out = (float*)d_out;
    float* y   = (float*)d_ws;                  // [64][196] channel sums

    float* out_idx   = out;                     // [64][6]  indices (as float)
    float* out_score = out + NBATCH * 6;        // [64][6]  scores
    float* all_sc    = out + 2 * NBATCH * 6;    // [64][837] all window scores

    k_chansum_wmma <<<dim3(NBATCH * NPTILE), dim3(256), 0, stream>>>(x, y);
    k_window_scores<<<dim3(NBATCH),          dim3(256), 0, stream>>>(y, all_sc);
    k_nms          <<<dim3(NBATCH * 3),      dim3(32),  0, stream>>>(all_sc, out_idx, out_score);
}